// gcn_layer_13271448945164
// MI455X (gfx1250) — compile-verified
//
#include <hip/hip_runtime.h>
#include <hip/hip_bf16.h>

#define N_NODES 50000
#define NFEAT   512
#define NHID    128
#define N_EDGES 800000
#define LN_EPS  1e-5f

typedef __attribute__((ext_vector_type(2))) float        v2f;
typedef __attribute__((ext_vector_type(8))) float        v8f;
typedef __attribute__((ext_vector_type(4))) unsigned int v4u;
typedef __attribute__((ext_vector_type(8))) int          v8i;
typedef __attribute__((ext_vector_type(4))) int          v4i;

// ---------------------------------------------------------------------------
// Kernel 1: degree accumulation (float atomics -> global_atomic_add_f32)
// ---------------------------------------------------------------------------
__global__ __launch_bounds__(256) void deg_kernel(const int* __restrict__ src,
                                                  const int* __restrict__ dst,
                                                  float* __restrict__ out_deg,
                                                  float* __restrict__ in_deg,
                                                  int n_edges) {
    int e = blockIdx.x * 256 + threadIdx.x;
    if (e >= n_edges) return;
    atomicAdd(out_deg + src[e], 1.0f);
    atomicAdd(in_deg + dst[e], 1.0f);
}

// ---------------------------------------------------------------------------
// Kernel 2: deg -> rsqrt(max(deg,1)) in place, both arrays back-to-back
// ---------------------------------------------------------------------------
__global__ __launch_bounds__(256) void norm_kernel(float* __restrict__ deg, int n) {
    int i = blockIdx.x * 256 + threadIdx.x;
    if (i >= n) return;
    deg[i] = rsqrtf(fmaxf(deg[i], 1.0f));
}

// ---------------------------------------------------------------------------
// Kernel 3: x = (h * out_norm[:,None]) @ W  via V_WMMA_F32_16X16X4_F32
// Block: 256 thr = 8 waves. Block owns a 16-row M-tile; wave w owns N-tile
// [16w, 16w+16). The raw 16x512 h-tile is DMA'd to LDS by the Tensor Data
// Mover (one tensor_load_to_lds per block, TENSORcnt-tracked); the per-row
// out_norm scale is applied at A-fragment read (VALU co-executes with WMMA).
//
// Fragment layouts (CDNA5 ISA 7.12.2, 32-bit):
//  A 16x4:  lanes 0-15: M=lane,   v0=K0, v1=K1 ; lanes 16-31: M=lane-16, v0=K2, v1=K3
//  B 4x16:  v0: lanes0-15 row K0, lanes16-31 row K1; v1: K2 / K3
//  C/D f32: VGPR i: lanes0-15 -> M=i, N=lane ; lanes16-31 -> M=i+8, N=lane-16
// ---------------------------------------------------------------------------
__global__ __launch_bounds__(256) void gemm_wmma_kernel(const float* __restrict__ h,
                                                        const float* __restrict__ Wm,
                                                        const float* __restrict__ out_norm,
                                                        float* __restrict__ x) {
    __shared__ float Atile[16][NFEAT];   // 32 KB, raw h tile

    const int m0  = blockIdx.x * 16;
    const int tid = threadIdx.x;

#if __has_builtin(__builtin_amdgcn_tensor_load_to_lds)
    // --- TDM path: one descriptor DMAs the whole 16x512 f32 tile to LDS ---
    if (tid < 32) {   // wave 0 issues; TDM ignores EXEC, issues once per wave
        const unsigned long long ga =
            (unsigned long long)(const void*)(h + (size_t)m0 * NFEAT);
        // Generic pointer to LDS: low 32 bits are the LDS byte address
        // (flat aperture rule).
        const unsigned lds = (unsigned)(unsigned long long)(void*)&Atile[0][0];

        v4u g0;                               // D# group 0 (ISA 8.3)
        g0.x = 1u;                            // count=1, user mode, no gather
        g0.y = lds;                           // lds_addr (bytes)
        g0.z = (unsigned)ga;                  // global_addr[31:0]
        g0.w = (unsigned)(ga >> 32) | (2u << 30); // global_addr[56:32] | type=2

        v8i g1;                               // D# group 1 (ISA 8.4)
        g1[0] = 0x00020000;                   // data_size=2 (4 bytes); no pad/iter/mcast
        g1[1] = (int)((unsigned)NFEAT << 16); // tensor_dim0[15:0]=512 (atomic_bar_addr=0)
        g1[2] = (int)((unsigned)N_NODES << 16); // tensor_dim0 hi=0 | tensor_dim1[15:0]=50000
        g1[3] = (int)((unsigned)NFEAT << 16); // tensor_dim1 hi=0 | tile_dim0=512
        g1[4] = 16;                           // tile_dim1=16, tile_dim2=0
        g1[5] = NFEAT;                        // tensor_dim0_stride[31:0]=512
        g1[6] = 0;                            // stride0 hi | dim1_stride lo (unused, 2D)
        g1[7] = 0;

        const v4i zz4 = {0, 0, 0, 0};         // groups 2/3 unused (<=2D tensor)
        const v8i zz8 = {0, 0, 0, 0, 0, 0, 0, 0};
        // 6-arg form (this toolchain: clang-23 / therock-10.0 headers)
        __builtin_amdgcn_tensor_load_to_lds(g0, g1, zz4, zz4, zz8, 0);
        __builtin_amdgcn_s_wait_tensorcnt(0); // TENSORcnt==0: tile is in LDS
    }
    __syncthreads();                          // publish LDS tile to all 8 waves
#else
    // --- Fallback: cooperative synchronous staging ---
    for (int i = tid; i < 16 * NFEAT; i += 256) {
        int r = i >> 9, c = i & (NFEAT - 1);
        Atile[r][c] = h[(size_t)(m0 + r) * NFEAT + c];
    }
    __syncthreads();
#endif

    const int wave = tid >> 5;       // 0..7 -> N-tile
    const int lane = tid & 31;
    const int n0   = wave * 16;
    const int row  = lane & 15;      // M (for A) / N (for B,D) within tile
    const bool hi  = lane >= 16;

    const float s = out_norm[m0 + row];   // per-row scale, applied at read

    v8f acc = {};
#pragma unroll 4
    for (int k0 = 0; k0 < NFEAT; k0 += 4) {
        // A fragment: two contiguous K values from LDS (ds_load_b64) * scale
        const int ka = k0 + (hi ? 2 : 0);
        v2f a;
        a.x = Atile[row][ka] * s;
        a.y = Atile[row][ka + 1] * s;
        // B fragment: W is [NFEAT][NHID] row-major; lane half selects K-row
        const int kb = k0 + (hi ? 1 : 0);
        v2f b;
        b.x = Wm[(size_t)kb       * NHID + n0 + row];
        b.y = Wm[(size_t)(kb + 2) * NHID + n0 + row];
        acc = __builtin_amdgcn_wmma_f32_16x16x4_f32(
            /*neg_a=*/false, a, /*neg_b=*/false, b,
            /*c_mod=*/(short)0, acc, /*reuse_a=*/false, /*reuse_b=*/false);
    }

#pragma unroll
    for (int i = 0; i < 8; ++i) {
        const int m = m0 + i + (hi ? 8 : 0);
        x[(size_t)m * NHID + n0 + row] = acc[i];
    }
}

// ---------------------------------------------------------------------------
// Kernel 4: agg[dst[e], :] += x[src[e], :]   (wave per edge, 4 f32/lane)
// ---------------------------------------------------------------------------
__global__ __launch_bounds__(256) void aggregate_kernel(const int* __restrict__ src,
                                                        const int* __restrict__ dst,
                                                        const float* __restrict__ x,
                                                        float* __restrict__ agg,
                                                        int n_edges) {
    const int e = blockIdx.x * 8 + (threadIdx.x >> 5);
    if (e >= n_edges) return;
    const int lane = threadIdx.x & 31;

    const float* xs = x + (size_t)src[e] * NHID;
    float* ad       = agg + (size_t)dst[e] * NHID;

    // prefetch next edge's source row into cache (global_prefetch_b8)
    if (e + 8 < n_edges)
        __builtin_prefetch(x + (size_t)src[e + 8] * NHID + lane * 4, 0, 3);

#pragma unroll
    for (int j = 0; j < 4; ++j) {
        const int c = lane + j * 32;
        atomicAdd(ad + c, xs[c]);
    }
}

// ---------------------------------------------------------------------------
// Kernel 5: in-place  y = LN(agg*in_norm + b)*gamma + beta  (wave per node)
// ---------------------------------------------------------------------------
__global__ __launch_bounds__(256) void finalize_ln_kernel(float* __restrict__ agg,
                                                          const float* __restrict__ in_norm,
                                                          const float* __restrict__ bias,
                                                          const float* __restrict__ gamma,
                                                          const float* __restrict__ beta,
                                                          int n_nodes) {
    const int node = blockIdx.x * 8 + (threadIdx.x >> 5);
    if (node >= n_nodes) return;
    const int lane = threadIdx.x & 31;

    const float s = in_norm[node];
    float* row = agg + (size_t)node * NHID;

    float v[4];
    float sum = 0.f;
#pragma unroll
    for (int j = 0; j < 4; ++j) {
        const int c = lane + j * 32;
        v[j] = row[c] * s + bias[c];
        sum += v[j];
    }
#pragma unroll
    for (int off = 16; off > 0; off >>= 1) sum += __shfl_xor(sum, off, 32);
    const float mu = sum * (1.0f / NHID);

    float var = 0.f;
#pragma unroll
    for (int j = 0; j < 4; ++j) { const float d = v[j] - mu; var += d * d; }
#pragma unroll
    for (int off = 16; off > 0; off >>= 1) var += __shfl_xor(var, off, 32);
    const float inv = rsqrtf(var * (1.0f / NHID) + LN_EPS);

#pragma unroll
    for (int j = 0; j < 4; ++j) {
        const int c = lane + j * 32;
        row[c] = (v[j] - mu) * inv * gamma[c] + beta[c];
    }
}

// ---------------------------------------------------------------------------
extern "C" void kernel_launch(void* const* d_in, const int* in_sizes, int n_in,
                              void* d_out, int out_size, void* d_ws, size_t ws_size,
                              hipStream_t stream) {
    const float* h     = (const float*)d_in[0];
    const int*   src   = (const int*)d_in[1];
    const int*   dst   = (const int*)d_in[2];
    const float* Wm    = (const float*)d_in[3];
    const float* bias  = (const float*)d_in[4];
    const float* gamma = (const float*)d_in[5];
    const float* beta  = (const float*)d_in[6];
    float* out = (float*)d_out;           // doubles as the agg accumulator

    float* out_deg = (float*)d_ws;                       // N_NODES
    float* in_deg  = out_deg + N_NODES;                  // N_NODES
    float* x       = in_deg + N_NODES;                   // N_NODES*NHID (offset 400000 B, 16B aligned)

    // zero degree counters + agg accumulator (capturable memset nodes)
    (void)hipMemsetAsync(d_ws, 0, 2 * (size_t)N_NODES * sizeof(float), stream);
    (void)hipMemsetAsync(d_out, 0, (size_t)out_size * sizeof(float), stream);

    deg_kernel<<<(N_EDGES + 255) / 256, 256, 0, stream>>>(src, dst, out_deg, in_deg, N_EDGES);
    norm_kernel<<<(2 * N_NODES + 255) / 256, 256, 0, stream>>>(out_deg, 2 * N_NODES);
    gemm_wmma_kernel<<<N_NODES / 16, 256, 0, stream>>>(h, Wm, out_deg, x);
    aggregate_kernel<<<(N_EDGES + 7) / 8, 256, 0, stream>>>(src, dst, x, out, N_EDGES);
    finalize_ln_kernel<<<(N_NODES + 7) / 8, 256, 0, stream>>>(out, in_deg, bias, gamma, beta, N_NODES);
}